// dMaSIFConv_seg_61847529062864
// MI455X (gfx1250) — compile-verified
//
#include <hip/hip_runtime.h>

// Fused 4-layer MLP (12 chained 64x64 GEMMs) for gfx1250 using fp32 WMMA.
// D = A(16x4 f32) x B(4x16 f32) + C(16x16 f32) via v_wmma_f32_16x16x4_f32.
// Each wave owns a 32-row tile: two A fragments share every B fragment
// (2 WMMAs per B ds_read_b64), halving weight-side LDS traffic per FLOP.

typedef float v2f __attribute__((ext_vector_type(2)));
typedef float v8f __attribute__((ext_vector_type(8)));

#define NPTS       500000
#define CCH        64
#define NSTAGE     12
#define TROWS      32                     // rows per wave tile
#define NTILES     (NPTS / TROWS)         // 15625 exact 32-row tiles
#define WAVES_PB   8
#define TPB        (WAVES_PB * 32)
#define WPITCH     80                     // float2 pitch per packed-k row (pad: +32 banks for +1 row)
#define XPITCH     68                     // float pitch per activation row (pad: 4-bank skew)
#define W_BYTES    (NSTAGE * 32 * WPITCH * 8)     // 245760
#define B_BYTES    (NSTAGE * CCH * 4)             // 3072
#define X_FLOATS   (TROWS * XPITCH)               // 2176 per wave
#define X_BYTES    (WAVES_PB * X_FLOATS * 4)      // 69632
#define SMEM_BYTES (W_BYTES + B_BYTES + X_BYTES)  // 318464 < 320KB

static __device__ __forceinline__ v8f splat8(float x) {
  v8f v = {x, x, x, x, x, x, x, x};
  return v;
}

__global__ __launch_bounds__(TPB) void dmasif_fused_mlp(
    const float* __restrict__ feat,
    const float* __restrict__ W0, const float* __restrict__ B0,
    const float* __restrict__ W1, const float* __restrict__ B1,
    const float* __restrict__ W2, const float* __restrict__ B2,
    float* __restrict__ out) {
  extern __shared__ char smem[];
  v2f*   wlds  = (v2f*)smem;                          // [12][32][WPITCH] packed B-layout weights
  float* blds  = (float*)(smem + W_BYTES);            // [12][64] biases
  float* xbase = (float*)(smem + W_BYTES + B_BYTES);  // [8][32][XPITCH] per-wave activation tiles

  const int tid  = threadIdx.x;
  const int lane = tid & 31;
  const int wave = tid >> 5;
  const int m    = lane & 15;  // A: row | B/D: column
  const int u    = lane >> 4;  // half-wave selector

  // ---- Stage all 12 weight matrices into LDS, pre-packed in B-operand layout ----
  // wp[s][kk][n] = ( W[2kk][n], W[2kk+1][n] )  so a B chunk is one ds_read_b64.
  for (int idx = tid; idx < NSTAGE * 32 * CCH; idx += TPB) {
    int s  = idx / (32 * CCH);
    int r  = idx - s * (32 * CCH);
    int kk = r >> 6;
    int n  = r & 63;
    int l  = s / 3;
    int wh = s - l * 3;
    const float* Wsrc = (wh == 0) ? W0 : (wh == 1) ? W1 : W2;
    Wsrc += (size_t)l * CCH * CCH;
    v2f w;
    w.x = Wsrc[(2 * kk + 0) * CCH + n];
    w.y = Wsrc[(2 * kk + 1) * CCH + n];
    wlds[(s * 32 + kk) * WPITCH + n] = w;
  }
  for (int idx = tid; idx < NSTAGE * CCH; idx += TPB) {
    int s  = idx >> 6;
    int l  = s / 3;
    int wh = s - l * 3;
    const float* Bsrc = (wh == 0) ? B0 : (wh == 1) ? B1 : B2;
    blds[idx] = Bsrc[l * CCH + (idx & 63)];
  }
  __syncthreads();

  float* xs = xbase + wave * X_FLOATS;

  // Each wave independently grid-strides over 32-row tiles (no intra-loop barriers).
  for (int t = blockIdx.x * WAVES_PB + wave; t < NTILES; t += gridDim.x * WAVES_PB) {
    // Coalesced global -> LDS load of the 32x64 fp32 tile (b128 per lane).
    const float* src = feat + (size_t)t * (TROWS * CCH);
#pragma unroll
    for (int i = 0; i < 16; ++i) {
      int e = lane + i * 32;
      int r = e >> 4;
      int c = (e & 15) << 2;
      *(float4*)(xs + r * XPITCH + c) = *(const float4*)(src + r * CCH + c);
    }
    asm volatile("s_wait_dscnt 0" ::: "memory");

    for (int s = 0; s < NSTAGE; ++s) {
      const float* bb = blds + s * CCH;
      float bv0 = bb[ 0 + m], bv1 = bb[16 + m], bv2 = bb[32 + m], bv3 = bb[48 + m];
      v8f acc0 = splat8(bv0);   // rows 0..15, cols  0..15
      v8f acc1 = splat8(bv1);   // rows 0..15, cols 16..31
      v8f acc2 = splat8(bv2);   // rows 0..15, cols 32..47
      v8f acc3 = splat8(bv3);   // rows 0..15, cols 48..63
      v8f acc4 = splat8(bv0);   // rows 16..31, cols  0..15
      v8f acc5 = splat8(bv1);
      v8f acc6 = splat8(bv2);
      v8f acc7 = splat8(bv3);
      const v2f* wst = wlds + (size_t)(s * 32) * WPITCH;
#pragma unroll
      for (int c = 0; c < 16; ++c) {
        // A chunks: lane m(+16u) holds row m (tile half 0/1), K = {4c+2u, 4c+2u+1}
        v2f a0 = *(const v2f*)(xs + (m     ) * XPITCH + 4 * c + 2 * u);
        v2f a1 = *(const v2f*)(xs + (m + 16) * XPITCH + 4 * c + 2 * u);
        // B chunks: lane holds col n0+m, K pair {4c+2u, 4c+2u+1}; shared by both A tiles
        const v2f* wrow = wst + (2 * c + u) * WPITCH + m;
        v2f b0 = wrow[ 0], b1 = wrow[16], b2 = wrow[32], b3 = wrow[48];
        acc0 = __builtin_amdgcn_wmma_f32_16x16x4_f32(false, a0, false, b0, (short)0, acc0, false, false);
        acc1 = __builtin_amdgcn_wmma_f32_16x16x4_f32(false, a0, false, b1, (short)0, acc1, false, false);
        acc2 = __builtin_amdgcn_wmma_f32_16x16x4_f32(false, a0, false, b2, (short)0, acc2, false, false);
        acc3 = __builtin_amdgcn_wmma_f32_16x16x4_f32(false, a0, false, b3, (short)0, acc3, false, false);
        acc4 = __builtin_amdgcn_wmma_f32_16x16x4_f32(false, a1, false, b0, (short)0, acc4, false, false);
        acc5 = __builtin_amdgcn_wmma_f32_16x16x4_f32(false, a1, false, b1, (short)0, acc5, false, false);
        acc6 = __builtin_amdgcn_wmma_f32_16x16x4_f32(false, a1, false, b2, (short)0, acc6, false, false);
        acc7 = __builtin_amdgcn_wmma_f32_16x16x4_f32(false, a1, false, b3, (short)0, acc7, false, false);
      }
      if ((s % 3) == 1) {  // linear_block ReLU stage
#pragma unroll
        for (int r = 0; r < 8; ++r) {
          acc0[r] = fmaxf(acc0[r], 0.0f);
          acc1[r] = fmaxf(acc1[r], 0.0f);
          acc2[r] = fmaxf(acc2[r], 0.0f);
          acc3[r] = fmaxf(acc3[r], 0.0f);
          acc4[r] = fmaxf(acc4[r], 0.0f);
          acc5[r] = fmaxf(acc5[r], 0.0f);
          acc6[r] = fmaxf(acc6[r], 0.0f);
          acc7[r] = fmaxf(acc7[r], 0.0f);
        }
      }
      // D (16x16 f32, vgpr r = row r+8u, col m) -> LDS row-major: transpose for next stage's A.
#pragma unroll
      for (int r = 0; r < 8; ++r) {
        float* row0 = xs + (r + 8 * u     ) * XPITCH + m;
        float* row1 = xs + (r + 8 * u + 16) * XPITCH + m;
        row0[ 0] = acc0[r];
        row0[16] = acc1[r];
        row0[32] = acc2[r];
        row0[48] = acc3[r];
        row1[ 0] = acc4[r];
        row1[16] = acc5[r];
        row1[32] = acc6[r];
        row1[48] = acc7[r];
      }
      asm volatile("s_wait_dscnt 0" ::: "memory");
    }

    // Coalesced LDS -> global store of the finished 32x64 tile.
    float* dst = out + (size_t)t * (TROWS * CCH);
#pragma unroll
    for (int i = 0; i < 16; ++i) {
      int e = lane + i * 32;
      int r = e >> 4;
      int c = (e & 15) << 2;
      *(float4*)(dst + r * CCH + c) = *(const float4*)(xs + r * XPITCH + c);
    }
  }
}

extern "C" void kernel_launch(void* const* d_in, const int* in_sizes, int n_in,
                              void* d_out, int out_size, void* d_ws, size_t ws_size,
                              hipStream_t stream) {
  (void)in_sizes; (void)n_in; (void)d_ws; (void)ws_size; (void)out_size;
  // setup_inputs order: features, points, nuv, W0, b0, W1, b1, W2, b2
  const float* feat = (const float*)d_in[0];
  const float* W0   = (const float*)d_in[3];
  const float* B0   = (const float*)d_in[4];
  const float* W1   = (const float*)d_in[5];
  const float* B1   = (const float*)d_in[6];
  const float* W2   = (const float*)d_in[7];
  const float* B2   = (const float*)d_in[8];
  float* out = (float*)d_out;

  hipFuncSetAttribute(reinterpret_cast<const void*>(dmasif_fused_mlp),
                      hipFuncAttributeMaxDynamicSharedMemorySize, SMEM_BYTES);
  dmasif_fused_mlp<<<dim3(512), dim3(TPB), SMEM_BYTES, stream>>>(
      feat, W0, B0, W1, B1, W2, B2, out);
}